// Decoder_66443144069664
// MI455X (gfx1250) — compile-verified
//
#include <hip/hip_runtime.h>
#include <hip/hip_bf16.h>
#include <math.h>

typedef __attribute__((ext_vector_type(2))) float v2f;
typedef __attribute__((ext_vector_type(8))) float v8f;

#define S1 16
#define S2 512
#define ROWS (S1 * S2)      // 8192
#define INT_DIM 50
#define ANOTHER 80
#define TOPICS 50
#define VOCAB 8000
#define KPAD 64             // padded K for p
#define BN_EPS 1e-5f

// -------------------------------------------------------------------------
// Kernel 1: out_notes = zV @ zW^T + bu @ bp^T + bu_ @ bp_^T   [16,512]
// -------------------------------------------------------------------------
__global__ void notes_kernel(const float* __restrict__ zV,
                             const float* __restrict__ zW,
                             const float* __restrict__ bu,
                             const float* __restrict__ bp,
                             const float* __restrict__ bu_,
                             const float* __restrict__ bp_,
                             float* __restrict__ out) {
    int idx = blockIdx.x * blockDim.x + threadIdx.x;
    if (idx >= ROWS) return;
    int i = idx >> 9;          // / 512
    int j = idx & 511;
    float acc = bu[i] * bp[j] + bu_[i] * bp_[j];
    #pragma unroll 10
    for (int k = 0; k < INT_DIM; ++k)
        acc += zV[i * INT_DIM + k] * zW[j * INT_DIM + k];
    out[idx] = acc;
}

// -------------------------------------------------------------------------
// Kernel 2: theta = relu([zV_i,zW_j] @ W1 + b1) @ W2 + b2 ; p = softmax(theta)
// One wave per row; p written zero-padded to KPAD columns.
// -------------------------------------------------------------------------
__global__ __launch_bounds__(256) void theta_softmax_kernel(
    const float* __restrict__ zV, const float* __restrict__ zW,
    const float* __restrict__ W1, const float* __restrict__ b1,
    const float* __restrict__ W2, const float* __restrict__ b2,
    float* __restrict__ p_pad) {
    __shared__ float sW1[2 * INT_DIM * ANOTHER];   // 100x80
    __shared__ float sb1[ANOTHER];
    __shared__ float sW2[ANOTHER * TOPICS];        // 80x50
    __shared__ float sb2[TOPICS];
    __shared__ float sfeat[8][2 * INT_DIM];
    __shared__ float sh[8][ANOTHER];

    const int tid = threadIdx.x;
    for (int t = tid; t < 2 * INT_DIM * ANOTHER; t += 256) sW1[t] = W1[t];
    for (int t = tid; t < ANOTHER * TOPICS; t += 256)      sW2[t] = W2[t];
    if (tid < ANOTHER) sb1[tid] = b1[tid];
    if (tid < TOPICS)  sb2[tid] = b2[tid];

    const int wv = tid >> 5;
    const int lane = tid & 31;
    const int row = blockIdx.x * 8 + wv;
    const int i = row >> 9;
    const int j = row & 511;

    if (lane < 25) {
        sfeat[wv][2 * lane]               = zV[i * INT_DIM + 2 * lane];
        sfeat[wv][2 * lane + 1]           = zV[i * INT_DIM + 2 * lane + 1];
        sfeat[wv][INT_DIM + 2 * lane]     = zW[j * INT_DIM + 2 * lane];
        sfeat[wv][INT_DIM + 2 * lane + 1] = zW[j * INT_DIM + 2 * lane + 1];
    }
    __syncthreads();

    // hidden layer: h[80] = relu(feat @ W1 + b1)
    for (int e = lane; e < ANOTHER; e += 32) {
        float acc = sb1[e];
        for (int k = 0; k < 2 * INT_DIM; ++k)
            acc += sfeat[wv][k] * sW1[k * ANOTHER + e];
        sh[wv][e] = fmaxf(acc, 0.0f);
    }
    __syncthreads();

    // theta: lane handles columns lane and lane+32
    float t0, t1 = -INFINITY;
    {
        float acc = sb2[lane];
        for (int k = 0; k < ANOTHER; ++k)
            acc += sh[wv][k] * sW2[k * TOPICS + lane];
        t0 = acc;
    }
    const int e1 = lane + 32;
    if (e1 < TOPICS) {
        float acc = sb2[e1];
        for (int k = 0; k < ANOTHER; ++k)
            acc += sh[wv][k] * sW2[k * TOPICS + e1];
        t1 = acc;
    }
    // wave-wide softmax over 50 entries
    float m = fmaxf(t0, t1);
    for (int off = 16; off > 0; off >>= 1) m = fmaxf(m, __shfl_xor(m, off, 32));
    float ex0 = __expf(t0 - m);
    float ex1 = (e1 < TOPICS) ? __expf(t1 - m) : 0.0f;
    float s = ex0 + ex1;
    for (int off = 16; off > 0; off >>= 1) s += __shfl_xor(s, off, 32);
    float inv = 1.0f / s;
    p_pad[row * KPAD + lane]      = ex0 * inv;                       // cols 0..31
    p_pad[row * KPAD + 32 + lane] = (e1 < TOPICS) ? ex1 * inv : 0.f; // cols 32..63
}

// -------------------------------------------------------------------------
// Kernel 3: logits = p @ W3 + b3   [8192, 8000] via V_WMMA_F32_16X16X4_F32
// One wave per 16x16 tile. 512 x 500 tiles, exact cover, EXEC all-ones.
// -------------------------------------------------------------------------
__global__ __launch_bounds__(256) void logits_wmma_kernel(
    const float* __restrict__ p,    // [8192, KPAD] zero-padded
    const float* __restrict__ W3,   // [50, 8000]
    const float* __restrict__ b3,   // [8000]
    float* __restrict__ out) {      // [8192, 8000]
    const int wave = (blockIdx.x * blockDim.x + threadIdx.x) >> 5;
    const int lane = threadIdx.x & 31;
    const int TILES_N = VOCAB / 16;          // 500
    const int tm = wave / TILES_N;
    const int tn = wave % TILES_N;
    const int m0 = tm * 16;
    const int n0 = tn * 16;
    const int half = lane >> 4;              // 0: K even pair, 1: K odd pair
    const int l = lane & 15;

    v8f acc = {};
    #pragma unroll
    for (int k0 = 0; k0 < 52; k0 += 4) {     // K = 0..51 (p cols 50,51 are zero)
        const int ka = k0 + 2 * half;
        // A: 16x4 f32, lane l row M=m0+l; VGPR0=K ka, VGPR1=K ka+1
        v2f a;
        a.x = p[(m0 + l) * KPAD + ka];
        a.y = p[(m0 + l) * KPAD + ka + 1];
        // B: 4x16 f32, row ka striped across lanes 0-15 / 16-31
        v2f b;
        b.x = (ka < INT_DIM)     ? W3[(size_t)ka * VOCAB + n0 + l]       : 0.0f;
        b.y = (ka + 1 < INT_DIM) ? W3[(size_t)(ka + 1) * VOCAB + n0 + l] : 0.0f;
        acc = __builtin_amdgcn_wmma_f32_16x16x4_f32(
            /*neg_a=*/false, a, /*neg_b=*/false, b,
            /*c_mod=*/(short)0, acc, /*reuse_a=*/false, /*reuse_b=*/false);
    }
    // C/D layout: VGPR g -> M = m0 + g + 8*half, N = n0 + l
    #pragma unroll
    for (int g = 0; g < 8; ++g) {
        const int mrow = m0 + g + 8 * half;
        const int ncol = n0 + l;
        out[(size_t)mrow * VOCAB + ncol] = acc[g] + b3[ncol];
    }
}

// -------------------------------------------------------------------------
// Kernel 4: per-column batch stats over 8192 rows -> mean, invstd
// -------------------------------------------------------------------------
__global__ void col_stats_kernel(const float* __restrict__ text,
                                 float* __restrict__ mean,
                                 float* __restrict__ invstd) {
    int c = blockIdx.x * blockDim.x + threadIdx.x;
    if (c >= VOCAB) return;
    float s = 0.0f, ss = 0.0f;
    for (int r = 0; r < ROWS; ++r) {
        float x = text[(size_t)r * VOCAB + c];
        s += x;
        ss += x * x;
    }
    const float inv_n = 1.0f / (float)ROWS;
    float mu = s * inv_n;
    float var = ss * inv_n - mu * mu;
    mean[c] = mu;
    invstd[c] = rsqrtf(var + BN_EPS);
}

// -------------------------------------------------------------------------
// Kernel 5: BN normalize + row softmax, in place. One block per row.
// -------------------------------------------------------------------------
__global__ __launch_bounds__(256) void bn_softmax_kernel(
    float* __restrict__ text,
    const float* __restrict__ mean,
    const float* __restrict__ invstd) {
    __shared__ float srow[VOCAB];
    __shared__ float sred[256];
    const int tid = threadIdx.x;
    float* row = text + (size_t)blockIdx.x * VOCAB;

    float lmax = -INFINITY;
    for (int e = tid; e < VOCAB; e += 256) {
        float z = (row[e] - mean[e]) * invstd[e];
        srow[e] = z;
        lmax = fmaxf(lmax, z);
    }
    sred[tid] = lmax;
    __syncthreads();
    for (int off = 128; off > 0; off >>= 1) {
        if (tid < off) sred[tid] = fmaxf(sred[tid], sred[tid + off]);
        __syncthreads();
    }
    const float m = sred[0];
    __syncthreads();

    float lsum = 0.0f;
    for (int e = tid; e < VOCAB; e += 256) {
        float ex = __expf(srow[e] - m);
        srow[e] = ex;
        lsum += ex;
    }
    sred[tid] = lsum;
    __syncthreads();
    for (int off = 128; off > 0; off >>= 1) {
        if (tid < off) sred[tid] += sred[tid + off];
        __syncthreads();
    }
    const float inv = 1.0f / sred[0];
    for (int e = tid; e < VOCAB; e += 256) row[e] = srow[e] * inv;
}

// -------------------------------------------------------------------------
extern "C" void kernel_launch(void* const* d_in, const int* in_sizes, int n_in,
                              void* d_out, int out_size, void* d_ws, size_t ws_size,
                              hipStream_t stream) {
    const float* zV  = (const float*)d_in[0];
    const float* zW  = (const float*)d_in[1];
    const float* bu  = (const float*)d_in[2];
    const float* bp  = (const float*)d_in[3];
    const float* bu_ = (const float*)d_in[4];
    const float* bp_ = (const float*)d_in[5];
    const float* W1  = (const float*)d_in[6];
    const float* b1  = (const float*)d_in[7];
    const float* W2  = (const float*)d_in[8];
    const float* b2  = (const float*)d_in[9];
    const float* W3  = (const float*)d_in[10];
    const float* b3  = (const float*)d_in[11];

    float* out_notes = (float*)d_out;              // 8192 floats
    float* out_text  = out_notes + ROWS;           // 8192*8000 floats

    float* p_pad  = (float*)d_ws;                  // [8192, 64]
    float* mean   = p_pad + (size_t)ROWS * KPAD;   // [8000]
    float* invstd = mean + VOCAB;                  // [8000]

    // 1) notes
    notes_kernel<<<(ROWS + 255) / 256, 256, 0, stream>>>(zV, zW, bu, bp, bu_, bp_, out_notes);

    // 2) theta + softmax -> p_pad (one wave per row, 8 waves/block)
    theta_softmax_kernel<<<ROWS / 8, 256, 0, stream>>>(zV, zW, W1, b1, W2, b2, p_pad);

    // 3) logits = p @ W3 + b3 (WMMA f32; 512*500 tiles, 8 waves/block)
    const int total_waves = (ROWS / 16) * (VOCAB / 16);     // 256000
    logits_wmma_kernel<<<total_waves / 8, 256, 0, stream>>>(p_pad, W3, b3, out_text);

    // 4) column stats
    col_stats_kernel<<<(VOCAB + 255) / 256, 256, 0, stream>>>(out_text, mean, invstd);

    // 5) batchnorm + row softmax (in place)
    bn_softmax_kernel<<<ROWS, 256, 0, stream>>>(out_text, mean, invstd);
}